// GATWorker_34892314312747
// MI455X (gfx1250) — compile-verified
//
#include <hip/hip_runtime.h>
#include <hip/hip_bf16.h>

#define N_NODES 50000
#define N_EDGES 800000
#define E_TOT   (N_EDGES + N_NODES)   // self loops appended
#define HEADS   4
#define FEAT    512                   // HEADS * 128
#define CH      128
#define NEG_SLOPE 0.2f

typedef __attribute__((ext_vector_type(2))) float v2f;
typedef __attribute__((ext_vector_type(8))) float v8f;

// ---------------------------------------------------------------------------
// GEMM: OUT[m][n] = sum_k X[m][k] * W[n][k]     X:[M,K]  W:[512,K]  OUT:[M,512]
// One wave computes a 16x64 tile via V_WMMA_F32_16X16X4_F32 (full fp32).
// M % 16 == 0, K % 4 == 0, N = 512 fixed.  EXEC is all-ones (wave-uniform exit).
// ---------------------------------------------------------------------------
__global__ __launch_bounds__(256) void gemm_xWT(const float* __restrict__ X,
                                                const float* __restrict__ W,
                                                float* __restrict__ OUT,
                                                int M, int K) {
  const int lane    = threadIdx.x & 31;
  const int waveId  = blockIdx.x * (blockDim.x >> 5) + (threadIdx.x >> 5);
  const int tilesN  = 8;                      // 512 / 64
  const int mTile   = waveId / tilesN;
  const int nTile   = waveId % tilesN;
  if (mTile * 16 >= M) return;                // wave-uniform

  const int row  = mTile * 16 + (lane & 15);  // A: M = lane%16
  const int kq   = (lane >> 4) * 2;           // A/B: K pair select per half-wave
  const int colB = nTile * 64 + (lane & 15);  // B: N = lane%16 within sub-tile

  v8f acc0 = {}, acc1 = {}, acc2 = {}, acc3 = {};
  const float* xr = X + (size_t)row * K;

  for (int kb = 0; kb < K; kb += 4) {
    v2f a;
    a.x = xr[kb + kq];
    a.y = xr[kb + kq + 1];
    v2f b0, b1, b2, b3;
    b0.x = W[(size_t)(colB +  0) * K + kb + kq];
    b0.y = W[(size_t)(colB +  0) * K + kb + kq + 1];
    b1.x = W[(size_t)(colB + 16) * K + kb + kq];
    b1.y = W[(size_t)(colB + 16) * K + kb + kq + 1];
    b2.x = W[(size_t)(colB + 32) * K + kb + kq];
    b2.y = W[(size_t)(colB + 32) * K + kb + kq + 1];
    b3.x = W[(size_t)(colB + 48) * K + kb + kq];
    b3.y = W[(size_t)(colB + 48) * K + kb + kq + 1];
    acc0 = __builtin_amdgcn_wmma_f32_16x16x4_f32(false, a, false, b0, (short)0, acc0, false, false);
    acc1 = __builtin_amdgcn_wmma_f32_16x16x4_f32(false, a, false, b1, (short)0, acc1, false, false);
    acc2 = __builtin_amdgcn_wmma_f32_16x16x4_f32(false, a, false, b2, (short)0, acc2, false, false);
    acc3 = __builtin_amdgcn_wmma_f32_16x16x4_f32(false, a, false, b3, (short)0, acc3, false, false);
  }

  // C/D layout: VGPR r -> lanes 0-15: (M = base+r, N = lane); lanes 16-31: (M = base+8+r)
  const int mBase = mTile * 16 + (lane >> 4) * 8;
  const int nIdx  = nTile * 64 + (lane & 15);
  v8f* accs[4] = {&acc0, &acc1, &acc2, &acc3};
  #pragma unroll
  for (int j = 0; j < 4; ++j) {
    #pragma unroll
    for (int r = 0; r < 8; ++r)
      OUT[(size_t)(mBase + r) * FEAT + nIdx + j * 16] = (*accs[j])[r];
  }
}

// ---------------------------------------------------------------------------
// Per-(node, head) attention logits: a_s/a_d [N,4]
// ---------------------------------------------------------------------------
__global__ void attn_coef(const float* __restrict__ xh,
                          const float* __restrict__ att_src,
                          const float* __restrict__ att_dst,
                          float* __restrict__ a_s, float* __restrict__ a_d, int n) {
  int t = blockIdx.x * blockDim.x + threadIdx.x;
  if (t >= n * HEADS) return;
  int node = t >> 2, h = t & 3;
  const float* v  = xh + (size_t)node * FEAT + h * CH;
  const float* ws = att_src + h * CH;
  const float* wd = att_dst + h * CH;
  float ss = 0.f, sd = 0.f;
  #pragma unroll 4
  for (int c = 0; c < CH; ++c) { float x = v[c]; ss += x * ws[c]; sd += x * wd[c]; }
  a_s[t] = ss; a_d[t] = sd;
}

// ---------------------------------------------------------------------------
// CSR construction (dst-sorted edge list, self loops appended)
// ---------------------------------------------------------------------------
__global__ void zero_i32(int* p, int n) {
  int i = blockIdx.x * blockDim.x + threadIdx.x;
  if (i < n) p[i] = 0;
}

__global__ void count_deg(const long long* __restrict__ ei, int* __restrict__ deg) {
  int e = blockIdx.x * blockDim.x + threadIdx.x;
  if (e >= E_TOT) return;
  int d = (e < N_EDGES) ? (int)ei[N_EDGES + e] : (e - N_EDGES);
  atomicAdd(&deg[d], 1);
}

__global__ __launch_bounds__(1024) void scan_deg(const int* __restrict__ deg,
                                                 int* __restrict__ row_ptr, int n) {
  __shared__ int sums[1024];
  const int tid = threadIdx.x;
  const int chunk = (n + 1023) / 1024;
  const int start = tid * chunk;
  int s = 0;
  for (int i = 0; i < chunk; ++i) { int idx = start + i; if (idx < n) s += deg[idx]; }
  sums[tid] = s;
  __syncthreads();
  for (int off = 1; off < 1024; off <<= 1) {
    int v = (tid >= off) ? sums[tid - off] : 0;
    __syncthreads();
    sums[tid] += v;
    __syncthreads();
  }
  int run = (tid == 0) ? 0 : sums[tid - 1];
  for (int i = 0; i < chunk; ++i) {
    int idx = start + i;
    if (idx < n) { row_ptr[idx] = run; run += deg[idx]; }
  }
  if (tid == 1023) row_ptr[n] = sums[1023];
}

__global__ void init_cursor(const int* __restrict__ row_ptr, int* __restrict__ cur, int n) {
  int i = blockIdx.x * blockDim.x + threadIdx.x;
  if (i < n) cur[i] = row_ptr[i];
}

__global__ void fill_csr(const long long* __restrict__ ei, int* __restrict__ cur,
                         int* __restrict__ col) {
  int e = blockIdx.x * blockDim.x + threadIdx.x;
  if (e >= E_TOT) return;
  int s, d;
  if (e < N_EDGES) { s = (int)ei[e]; d = (int)ei[N_EDGES + e]; }
  else             { s = d = e - N_EDGES; }
  int pos = atomicAdd(&cur[d], 1);
  col[pos] = s;
}

// ---------------------------------------------------------------------------
// Aggregation: one wave per destination node. Softmax over in-edges per head,
// weighted sum of xh[src], head-mean via wave32 shuffles. out:[N,128]
// ---------------------------------------------------------------------------
__global__ __launch_bounds__(128) void aggregate(const float* __restrict__ xh,
                                                 const float* __restrict__ a_s,
                                                 const float* __restrict__ a_d,
                                                 const int* __restrict__ row_ptr,
                                                 const int* __restrict__ col,
                                                 const float* __restrict__ bias,
                                                 float* __restrict__ out, int nN) {
  const int lane = threadIdx.x & 31;
  const int node = blockIdx.x * (blockDim.x >> 5) + (threadIdx.x >> 5);
  if (node >= nN) return;
  const int beg = row_ptr[node];
  const int end = row_ptr[node + 1];

  float ad[HEADS];
  #pragma unroll
  for (int h = 0; h < HEADS; ++h) ad[h] = a_d[node * HEADS + h];

  // pass 1: per-head max of leaky-relu(alpha), edges striped across lanes
  float mx[HEADS];
  #pragma unroll
  for (int h = 0; h < HEADS; ++h) mx[h] = -3.0e38f;
  for (int e = beg + lane; e < end; e += 32) {
    int s = col[e];
    #pragma unroll
    for (int h = 0; h < HEADS; ++h) {
      float al = a_s[s * HEADS + h] + ad[h];
      al = (al >= 0.f) ? al : NEG_SLOPE * al;
      mx[h] = fmaxf(mx[h], al);
    }
  }
  #pragma unroll
  for (int h = 0; h < HEADS; ++h) {
    #pragma unroll
    for (int m = 16; m >= 1; m >>= 1)
      mx[h] = fmaxf(mx[h], __shfl_xor(mx[h], m, 32));
    if (mx[h] < -1.0e37f) mx[h] = 0.f;   // isfinite guard (empty segment)
  }

  // pass 2: lane owns 16 contiguous channels (head = lane/8)
  const int myHead = lane >> 3;
  const int cbase  = lane * 16;
  float denom[HEADS] = {0.f, 0.f, 0.f, 0.f};
  float acc[16];
  #pragma unroll
  for (int k = 0; k < 16; ++k) acc[k] = 0.f;

  for (int e = beg; e < end; ++e) {
    int s = col[e];
    float eh[HEADS];
    #pragma unroll
    for (int h = 0; h < HEADS; ++h) {
      float al = a_s[s * HEADS + h] + ad[h];
      al = (al >= 0.f) ? al : NEG_SLOPE * al;
      eh[h] = __expf(al - mx[h]);
      denom[h] += eh[h];
    }
    const float w = eh[myHead];
    const float4* p = (const float4*)(xh + (size_t)s * FEAT + cbase);
    #pragma unroll
    for (int q = 0; q < 4; ++q) {
      float4 v = p[q];
      acc[q * 4 + 0] += w * v.x;
      acc[q * 4 + 1] += w * v.y;
      acc[q * 4 + 2] += w * v.z;
      acc[q * 4 + 3] += w * v.w;
    }
  }

  const float inv = 1.0f / (denom[myHead] + 1e-16f);
  #pragma unroll
  for (int k = 0; k < 16; ++k) {
    float v = acc[k] * inv;
    v += __shfl_xor(v, 8, 32);     // sum heads {h, h^1}
    v += __shfl_xor(v, 16, 32);    // sum heads {.., h^2}
    acc[k] = v;
  }
  if (lane < 8) {
    const int c0 = lane * 16;       // within-head channel offset
    #pragma unroll
    for (int k = 0; k < 16; ++k)
      out[(size_t)node * CH + c0 + k] = acc[k] * 0.25f + bias[c0 + k];
  }
}

// ---------------------------------------------------------------------------
extern "C" void kernel_launch(void* const* d_in, const int* in_sizes, int n_in,
                              void* d_out, int out_size, void* d_ws, size_t ws_size,
                              hipStream_t stream) {
  const float*     x        = (const float*)d_in[0];
  const long long* ei       = (const long long*)d_in[1];   // int64 [2, E]
  const float*     W1       = (const float*)d_in[2];
  const float*     att_src1 = (const float*)d_in[3];
  const float*     att_dst1 = (const float*)d_in[4];
  const float*     b1       = (const float*)d_in[5];
  const float*     W2       = (const float*)d_in[6];
  const float*     att_src2 = (const float*)d_in[7];
  const float*     att_dst2 = (const float*)d_in[8];
  const float*     b2       = (const float*)d_in[9];
  float* outp = (float*)d_out;

  // bump allocator over workspace (512B aligned)
  char*  ws  = (char*)d_ws;
  size_t off = 0;
  auto alloc = [&](size_t bytes) -> void* {
    void* p = ws + off;
    off += (bytes + 511) & ~(size_t)511;
    return p;
  };
  int*   deg     = (int*)alloc(sizeof(int) * N_NODES);        // reused as cursor
  int*   row_ptr = (int*)alloc(sizeof(int) * (N_NODES + 1));
  int*   col     = (int*)alloc(sizeof(int) * E_TOT);
  float* a_s     = (float*)alloc(sizeof(float) * N_NODES * HEADS);
  float* a_d     = (float*)alloc(sizeof(float) * N_NODES * HEADS);
  float* h1      = (float*)alloc(sizeof(float) * N_NODES * CH);    // layer-1 output
  float* xh      = (float*)alloc(sizeof(float) * (size_t)N_NODES * FEAT); // shared

  // ---- CSR build (dst-sorted, once; reused by both layers) ----
  zero_i32<<<(N_NODES + 255) / 256, 256, 0, stream>>>(deg, N_NODES);
  count_deg<<<(E_TOT + 255) / 256, 256, 0, stream>>>(ei, deg);
  scan_deg<<<1, 1024, 0, stream>>>(deg, row_ptr, N_NODES);
  init_cursor<<<(N_NODES + 255) / 256, 256, 0, stream>>>(row_ptr, deg, N_NODES);
  fill_csr<<<(E_TOT + 255) / 256, 256, 0, stream>>>(ei, deg, col);

  // ---- layer 1 ----
  {
    const int waves = (N_NODES / 16) * 8;             // 25000 waves, 8/block
    gemm_xWT<<<waves / 8, 256, 0, stream>>>(x, W1, xh, N_NODES, 256);
    attn_coef<<<(N_NODES * HEADS + 255) / 256, 256, 0, stream>>>(xh, att_src1, att_dst1, a_s, a_d, N_NODES);
    aggregate<<<(N_NODES + 3) / 4, 128, 0, stream>>>(xh, a_s, a_d, row_ptr, col, b1, h1, N_NODES);
  }
  // ---- layer 2 ----
  {
    const int waves = (N_NODES / 16) * 8;
    gemm_xWT<<<waves / 8, 256, 0, stream>>>(h1, W2, xh, N_NODES, 128);
    attn_coef<<<(N_NODES * HEADS + 255) / 256, 256, 0, stream>>>(xh, att_src2, att_dst2, a_s, a_d, N_NODES);
    aggregate<<<(N_NODES + 3) / 4, 128, 0, stream>>>(xh, a_s, a_d, row_ptr, col, b2, outp, N_NODES);
  }
}